// Decoder_30382598652307
// MI455X (gfx1250) — compile-verified
//
#include <hip/hip_runtime.h>
#include <float.h>
#include <math.h>

typedef __attribute__((ext_vector_type(2))) float v2f;
typedef __attribute__((ext_vector_type(8))) float v8f;

#define WMMA_F32(a, b, c) \
    __builtin_amdgcn_wmma_f32_16x16x4_f32(false, (a), false, (b), (short)0, (c), false, false)

// ---------------------------------------------------------------------------
// WMMA fp32 GEMM:  out[M,N] = act(A[M,K] @ W[K,N] + bias) (+ res)
// Each wave computes a 16x32 output tile (2 N-tiles, A fragment reused).
// Fast path (interior tile, uniform per block): no guards -> A loads merge to
// global_load_b64; slow path handles M/N/K edges.
// ---------------------------------------------------------------------------
__global__ void gemm_kernel(const float* __restrict__ A, const float* __restrict__ W,
                            const float* __restrict__ bias, const float* __restrict__ res,
                            float* __restrict__ out, int M, int N, int K, int relu)
{
    int lane = threadIdx.x;          // 32 threads = 1 wave
    int half = lane >> 4, r = lane & 15;
    int n0 = blockIdx.x * 32, m0 = blockIdx.y * 16;
    int col0 = n0 + r, col1 = n0 + 16 + r;
    int K4 = K & ~3;
    v8f c0 = {}, c1 = {};
    bool fast = (m0 + 16 <= M) && (n0 + 32 <= N);   // uniform per block

    if (fast) {
        const float* Arow = A + (size_t)(m0 + r) * K;
        for (int k0 = 0; k0 < K4; k0 += 4) {
            int ka = k0 + 2 * half;
            v2f a, b0, b1;
            a.x = Arow[ka];
            a.y = Arow[ka + 1];
            const float* w0 = W + (size_t)ka * N;
            b0.x = w0[col0];       b1.x = w0[col1];
            b0.y = w0[N + col0];   b1.y = w0[N + col1];
            c0 = WMMA_F32(a, b0, c0);
            c1 = WMMA_F32(a, b1, c1);
        }
        if (K4 < K) {              // K tail (K = 261, 517, ...)
            int ka = K4 + 2 * half;
            v2f a, b0, b1;
            a.x = (ka     < K) ? Arow[ka]     : 0.f;
            a.y = (ka + 1 < K) ? Arow[ka + 1] : 0.f;
            b0.x = (ka     < K) ? W[(size_t)ka * N + col0]       : 0.f;
            b0.y = (ka + 1 < K) ? W[(size_t)(ka + 1) * N + col0] : 0.f;
            b1.x = (ka     < K) ? W[(size_t)ka * N + col1]       : 0.f;
            b1.y = (ka + 1 < K) ? W[(size_t)(ka + 1) * N + col1] : 0.f;
            c0 = WMMA_F32(a, b0, c0);
            c1 = WMMA_F32(a, b1, c1);
        }
        float bv0 = bias ? bias[col0] : 0.f;
        float bv1 = bias ? bias[col1] : 0.f;
#pragma unroll
        for (int v = 0; v < 8; ++v) {
            int rr = m0 + v + 8 * half;
            float v0 = c0[v] + bv0, v1 = c1[v] + bv1;
            if (relu) { v0 = fmaxf(v0, 0.f); v1 = fmaxf(v1, 0.f); }
            if (res) {
                v0 += res[(size_t)rr * N + col0];
                v1 += res[(size_t)rr * N + col1];
            }
            out[(size_t)rr * N + col0] = v0;
            out[(size_t)rr * N + col1] = v1;
        }
    } else {
        int arow = m0 + r;
        bool rowOk = arow < M;
        for (int k0 = 0; k0 < K; k0 += 4) {
            int ka = k0 + 2 * half;
            v2f a, b0, b1;
            a.x = (rowOk && ka     < K) ? A[(size_t)arow * K + ka]     : 0.f;
            a.y = (rowOk && ka + 1 < K) ? A[(size_t)arow * K + ka + 1] : 0.f;
            b0.x = (ka     < K && col0 < N) ? W[(size_t)ka * N + col0]       : 0.f;
            b0.y = (ka + 1 < K && col0 < N) ? W[(size_t)(ka + 1) * N + col0] : 0.f;
            b1.x = (ka     < K && col1 < N) ? W[(size_t)ka * N + col1]       : 0.f;
            b1.y = (ka + 1 < K && col1 < N) ? W[(size_t)(ka + 1) * N + col1] : 0.f;
            c0 = WMMA_F32(a, b0, c0);
            c1 = WMMA_F32(a, b1, c1);
        }
        float bv0 = (bias && col0 < N) ? bias[col0] : 0.f;
        float bv1 = (bias && col1 < N) ? bias[col1] : 0.f;
#pragma unroll
        for (int v = 0; v < 8; ++v) {
            int rr = m0 + v + 8 * half;
            if (rr < M) {
                if (col0 < N) {
                    float val = c0[v] + bv0;
                    if (relu) val = fmaxf(val, 0.f);
                    if (res) val += res[(size_t)rr * N + col0];
                    out[(size_t)rr * N + col0] = val;
                }
                if (col1 < N) {
                    float val = c1[v] + bv1;
                    if (relu) val = fmaxf(val, 0.f);
                    if (res) val += res[(size_t)rr * N + col1];
                    out[(size_t)rr * N + col1] = val;
                }
            }
        }
    }
}

// ---------------------------------------------------------------------------
// Small-N GEMV (N <= 8): wave per output row. Lanes stride K (coalesced A
// reads), per-lane FMA into <=8 accumulators, shfl_xor tree reduce, lane 0
// stores. Used for the pc heads (N=3) and occ logits (N=8) where a 16x16 WMMA
// tile is mostly wasted and edge guards dominate.
// ---------------------------------------------------------------------------
__global__ void gemm_smalln_kernel(const float* __restrict__ A, const float* __restrict__ W,
                                   const float* __restrict__ bias, const float* __restrict__ res,
                                   float* __restrict__ out, int M, int N, int K, int relu)
{
    int wave = threadIdx.x >> 5;
    int lane = threadIdx.x & 31;
    int m = blockIdx.x * 8 + wave;
    if (m >= M) return;
    const float* Arow = A + (size_t)m * K;
    float acc[8];
#pragma unroll
    for (int n = 0; n < 8; ++n) acc[n] = 0.f;
    for (int k = lane; k < K; k += 32) {
        float av = Arow[k];
        const float* wr = W + (size_t)k * N;
#pragma unroll
        for (int n = 0; n < 8; ++n)
            if (n < N) acc[n] = fmaf(av, wr[n], acc[n]);
    }
#pragma unroll
    for (int n = 0; n < 8; ++n) {
#pragma unroll
        for (int o = 16; o > 0; o >>= 1) acc[n] += __shfl_xor(acc[n], o, 32);
    }
    if (lane == 0) {
#pragma unroll
        for (int n = 0; n < 8; ++n) {
            if (n < N) {
                float val = acc[n] + (bias ? bias[n] : 0.f);
                if (relu) val = fmaxf(val, 0.f);
                if (res) val += res[(size_t)m * N + n];
                out[(size_t)m * N + n] = val;
            }
        }
    }
}

// ---------------------------------------------------------------------------
// attn0 GEMM with composed A:  A[row,k] = kfeat[b,idx[row],k] - q[b,nq,k] + pos[row,k]
// row = (b*Np + nq)*16 + kk.  M = B*Np*16 (mult of 16), K = 256, Nout = 64:
// fully unguarded; row pointers hoisted out of the K loop.
// ---------------------------------------------------------------------------
__global__ void gemm_attn0_kernel(const float* __restrict__ W, const float* __restrict__ bias,
                                  float* __restrict__ out, int Np, int K, int Nout,
                                  const float* __restrict__ kfeat, const float* __restrict__ q,
                                  const float* __restrict__ pos, const int* __restrict__ idx)
{
    int lane = threadIdx.x;
    int half = lane >> 4, r = lane & 15;
    int n0 = blockIdx.x * 16, m0 = blockIdx.y * 16;
    int row = m0 + r;
    int b  = row / (Np * 16);
    int nq = (row >> 4) - b * Np;
    int nb = idx[row];
    const float* krow = kfeat + ((size_t)b * Np + nb) * K;
    const float* qrow = q     + ((size_t)b * Np + nq) * K;
    const float* prow = pos   + (size_t)row * K;
    int col = n0 + r;
    v8f c = {};
    for (int k0 = 0; k0 < K; k0 += 4) {
        int ka = k0 + 2 * half;
        v2f a, bb;
        a.x = krow[ka]     - qrow[ka]     + prow[ka];
        a.y = krow[ka + 1] - qrow[ka + 1] + prow[ka + 1];
        const float* w0 = W + (size_t)ka * Nout;
        bb.x = w0[col];
        bb.y = w0[Nout + col];
        c = WMMA_F32(a, bb, c);
    }
    float bv = bias[col];
#pragma unroll
    for (int v = 0; v < 8; ++v)
        out[(size_t)(m0 + v + 8 * half) * Nout + col] = c[v] + bv;
}

// ---------------------------------------------------------------------------
// Fused decoder_grp: per (b,n) block -- gather 16 neighbors [pcdiff|x] into LDS,
// h1 = relu(A@W1+b1) (half at a time in LDS), h2 accum = h1@W2 (WMMA), max over
// the 16 neighbors (== the WMMA M dimension) via accumulators + shfl_xor(16).
// ---------------------------------------------------------------------------
__global__ void grp_kernel(const float* __restrict__ x, const float* __restrict__ pc,
                           const int* __restrict__ idx,
                           const float* __restrict__ W1, const float* __restrict__ b1,
                           const float* __restrict__ W2, const float* __restrict__ b2,
                           float* __restrict__ out, int N, int C)
{
    extern __shared__ float smem[];
    int KA = C + 3;
    int KApad = (KA + 3) & ~3;
    int CH = C >> 1;
    float* As = smem;              // [16][KApad]
    float* Hs = smem + 16 * KApad; // [16][CH]
    int bn = blockIdx.x;
    int b = bn / N;
    int tid = threadIdx.x;         // 256 = 8 waves
    int wave = tid >> 5, lane = tid & 31;
    int half = lane >> 4, r = lane & 15;
    const int* id16 = idx + (size_t)bn * 16;
    for (int e = tid; e < 16 * KApad; e += 256) {
        int kk = e / KApad, c = e - kk * KApad;
        float v = 0.f;
        if (c < KA) {
            int nb = id16[kk];
            if (c < 3) v = pc[((size_t)b * N + nb) * 3 + c] - pc[(size_t)bn * 3 + c];
            else       v = x[((size_t)b * N + nb) * C + (c - 3)];
        }
        As[e] = v;
    }
    __syncthreads();
    int nt1 = CH >> 4;
    int nt2 = C >> 4;
    v8f zero = {};
    v8f acc[4];
#pragma unroll
    for (int j = 0; j < 4; ++j) acc[j] = zero;

    for (int hf = 0; hf < 2; ++hf) {
        int cbase = hf * CH;
        for (int t1 = wave; t1 < nt1; t1 += 8) {
            int n0 = t1 * 16;
            v8f cc = {};
            for (int k0 = 0; k0 < KApad; k0 += 4) {
                int ka = k0 + 2 * half;
                v2f a, bb;
                a.x = As[r * KApad + ka];
                a.y = As[r * KApad + ka + 1];
                int col = cbase + n0 + r;
                bb.x = (ka     < KA) ? W1[(size_t)ka * C + col]       : 0.f;
                bb.y = (ka + 1 < KA) ? W1[(size_t)(ka + 1) * C + col] : 0.f;
                cc = WMMA_F32(a, bb, cc);
            }
            float bv = b1[cbase + n0 + r];
#pragma unroll
            for (int v = 0; v < 8; ++v)
                Hs[(v + 8 * half) * CH + n0 + r] = fmaxf(cc[v] + bv, 0.f);
        }
        __syncthreads();
#pragma unroll
        for (int j = 0; j < 4; ++j) {
            int t2 = wave + 8 * j;
            if (t2 < nt2) {
                int col = t2 * 16 + r;
                for (int k0 = 0; k0 < CH; k0 += 4) {
                    int ka = k0 + 2 * half;
                    v2f a, bb;
                    a.x = Hs[r * CH + ka];
                    a.y = Hs[r * CH + ka + 1];
                    bb.x = W2[(size_t)(cbase + ka) * C + col];
                    bb.y = W2[(size_t)(cbase + ka + 1) * C + col];
                    acc[j] = WMMA_F32(a, bb, acc[j]);
                }
            }
        }
        __syncthreads();
    }
#pragma unroll
    for (int j = 0; j < 4; ++j) {
        int t2 = wave + 8 * j;
        if (t2 < nt2) {
            int col = t2 * 16 + r;
            float bv = b2[col];
            float mx = -FLT_MAX;
#pragma unroll
            for (int v = 0; v < 8; ++v) mx = fmaxf(mx, acc[j][v] + bv);
            mx = fmaxf(mx, __shfl_xor(mx, 16, 32));
            if (half == 0) out[(size_t)bn * C + col] = mx;
        }
    }
}

// --------------------------------- FPS -------------------------------------
__global__ void fps_kernel(const float* __restrict__ pc, int* __restrict__ out,
                           float* __restrict__ dists, int Nc, int S)
{
    int b = blockIdx.x;
    const float* p = pc + (size_t)b * Nc * 3;
    float* dd = dists + (size_t)b * Nc;
    __shared__ float sval[256];
    __shared__ int sidx[256];
    __shared__ int slast;
    int tid = threadIdx.x;
    for (int i = tid; i < Nc; i += 256) dd[i] = 1e10f;
    if (tid == 0) { out[(size_t)b * S] = 0; slast = 0; }
    __syncthreads();
    for (int s = 1; s < S; ++s) {
        int last = slast;
        float lx = p[last * 3], ly = p[last * 3 + 1], lz = p[last * 3 + 2];
        float bv = -1.f; int bi = 0x7fffffff;
        for (int i = tid; i < Nc; i += 256) {
            float dx = p[i * 3] - lx, dy = p[i * 3 + 1] - ly, dz = p[i * 3 + 2] - lz;
            float d = dx * dx + dy * dy + dz * dz;
            float nv = fminf(dd[i], d);
            dd[i] = nv;
            if (nv > bv || (nv == bv && i < bi)) { bv = nv; bi = i; }
        }
        sval[tid] = bv; sidx[tid] = bi;
        __syncthreads();
        for (int o = 128; o > 0; o >>= 1) {
            if (tid < o) {
                float ov = sval[tid + o]; int oi = sidx[tid + o];
                if (ov > sval[tid] || (ov == sval[tid] && oi < sidx[tid])) { sval[tid] = ov; sidx[tid] = oi; }
            }
            __syncthreads();
        }
        if (tid == 0) { out[(size_t)b * S + s] = sidx[0]; slast = sidx[0]; }
        __syncthreads();
    }
}

// ---------------------------------- kNN ------------------------------------
__global__ void knn_kernel(const float* __restrict__ qpc, const float* __restrict__ cpc,
                           int* __restrict__ outidx, int B, int Nq, int Nc)
{
    size_t t = (size_t)blockIdx.x * blockDim.x + threadIdx.x;
    if (t >= (size_t)B * Nq) return;
    int b = (int)(t / Nq);
    const float* qp = qpc + t * 3;
    const float* cb = cpc + (size_t)b * Nc * 3;
    float qx = qp[0], qy = qp[1], qz = qp[2];
    float bd[16]; int bi[16];
#pragma unroll
    for (int j = 0; j < 16; ++j) { bd[j] = FLT_MAX; bi[j] = 0; }
    for (int i = 0; i < Nc; ++i) {
        float dx = qx - cb[i * 3], dy = qy - cb[i * 3 + 1], dz = qz - cb[i * 3 + 2];
        float d = dx * dx + dy * dy + dz * dz;
        if (d < bd[15]) {
            int j = 15;
            while (j > 0 && bd[j - 1] > d) { bd[j] = bd[j - 1]; bi[j] = bi[j - 1]; --j; }
            bd[j] = d; bi[j] = i;
        }
    }
#pragma unroll
    for (int j = 0; j < 16; ++j) outidx[t * 16 + j] = bi[j];
}

// ------------------------- elementwise / utility ---------------------------
__global__ void copy_rows_kernel(const float* __restrict__ src, float* __restrict__ dst,
                                 int Nsrc, int Ndst, int rowOff, int C, size_t total)
{
    size_t t = (size_t)blockIdx.x * blockDim.x + threadIdx.x;
    if (t >= total) return;
    int c = (int)(t % C); size_t row = t / C;
    int i = (int)(row % Nsrc); int b = (int)(row / Nsrc);
    dst[((size_t)b * Ndst + rowOff + i) * C + c] = src[t];
}

__global__ void gather_rows_kernel(const float* __restrict__ src, const int* __restrict__ fidx,
                                   float* __restrict__ dst, int Nsrc, int S, int C, size_t total)
{
    size_t t = (size_t)blockIdx.x * blockDim.x + threadIdx.x;
    if (t >= total) return;
    int c = (int)(t % C); size_t row = t / C;
    int s = (int)(row % S); int b = (int)(row / S);
    dst[t] = src[((size_t)b * Nsrc + fidx[(size_t)b * S + s]) * C + c];
}

__global__ void pcdiff_kernel(const float* __restrict__ qpc, const float* __restrict__ cpc,
                              const int* __restrict__ idx, float* __restrict__ out,
                              int N, int Ns, size_t total)
{
    size_t t = (size_t)blockIdx.x * blockDim.x + threadIdx.x;
    if (t >= total) return;
    int c = (int)(t % 3); size_t m = t / 3; size_t bn = m >> 4;
    int b = (int)(bn / N);
    int nb = idx[m];
    out[t] = cpc[((size_t)b * Ns + nb) * 3 + c] - qpc[bn * 3 + c];
}

__global__ void bn_stats_kernel(const float* __restrict__ x, int M, int C, float* __restrict__ stats)
{
    int c = blockIdx.x;
    __shared__ float s1[256], s2[256];
    float a = 0.f, b = 0.f;
    for (int m = threadIdx.x; m < M; m += 256) {
        float v = x[(size_t)m * C + c]; a += v; b += v * v;
    }
    s1[threadIdx.x] = a; s2[threadIdx.x] = b;
    __syncthreads();
    for (int o = 128; o > 0; o >>= 1) {
        if (threadIdx.x < o) { s1[threadIdx.x] += s1[threadIdx.x + o]; s2[threadIdx.x] += s2[threadIdx.x + o]; }
        __syncthreads();
    }
    if (threadIdx.x == 0) {
        float mean = s1[0] / M;
        float var = s2[0] / M - mean * mean;
        stats[c] = mean; stats[C + c] = rsqrtf(var + 1e-5f);
    }
}

__global__ void bn_apply_relu_kernel(float* __restrict__ x, int C, const float* __restrict__ stats,
                                     const float* __restrict__ g, const float* __restrict__ b, size_t total)
{
    size_t t = (size_t)blockIdx.x * blockDim.x + threadIdx.x;
    if (t >= total) return;
    int c = (int)(t % C);
    float v = (x[t] - stats[c]) * stats[C + c] * g[c] + b[c];
    x[t] = fmaxf(v, 0.f);
}

__global__ void softmax_k_kernel(float* __restrict__ x, int C, size_t total /* = B*N*C */)
{
    size_t t = (size_t)blockIdx.x * blockDim.x + threadIdx.x;
    if (t >= total) return;
    int c = (int)(t % C); size_t bn = t / C;
    float* p = x + bn * 16 * C + c;
    float mx = -FLT_MAX;
    for (int k = 0; k < 16; ++k) mx = fmaxf(mx, p[(size_t)k * C]);
    float e[16], s = 0.f;
    for (int k = 0; k < 16; ++k) { e[k] = expf(p[(size_t)k * C] - mx); s += e[k]; }
    float inv = 1.f / s;
    for (int k = 0; k < 16; ++k) p[(size_t)k * C] = e[k] * inv;
}

__global__ void agg_kernel(const float* __restrict__ v, const float* __restrict__ pos,
                           const float* __restrict__ attn, const int* __restrict__ idx,
                           float* __restrict__ out, int N, int C, size_t total /* = B*N*C */)
{
    size_t t = (size_t)blockIdx.x * blockDim.x + threadIdx.x;
    if (t >= total) return;
    int c = (int)(t % C); size_t bn = t / C;
    int b = (int)(bn / N);
    float s = 0.f;
    for (int kk = 0; kk < 16; ++kk) {
        size_t m = bn * 16 + kk;
        int nb = idx[m];
        float vv = v[((size_t)b * N + nb) * C + c] + pos[m * C + c];
        s += vv * attn[m * C + c];
    }
    out[t] = s;
}

__global__ void repeat3_kernel(const float* __restrict__ pc, float* __restrict__ out,
                               int Nin, int G, size_t total)
{
    size_t t = (size_t)blockIdx.x * blockDim.x + threadIdx.x;
    if (t >= total) return;
    int c = (int)(t % 3); size_t row = t / 3;
    int i = (int)(row % (Nin * G)); int b = (int)(row / (Nin * G));
    out[t] = pc[((size_t)b * Nin + i / G) * 3 + c];
}

__global__ void build_feat_kernel(const float* __restrict__ coarse, const float* __restrict__ gvec,
                                  float* __restrict__ out, int N, int Cg, float gs, size_t total)
{
    size_t t = (size_t)blockIdx.x * blockDim.x + threadIdx.x;
    if (t >= total) return;
    int Cin = 5 + Cg;
    int col = (int)(t % Cin); size_t row = t / Cin;
    int i = (int)(row % N); int b = (int)(row / N);
    float v;
    if (col < 2) { int gi = i & 3; v = (col == 0) ? ((gi >> 1) ? gs : -gs) : ((gi & 1) ? gs : -gs); }
    else if (col < 5) v = coarse[row * 3 + (col - 2)];
    else v = gvec[(size_t)b * Cg + (col - 5)];
    out[t] = v;
}

// ------------------------------ region group -------------------------------
__global__ void occ_top1_kernel(const float* __restrict__ logits, int M,
                                float* __restrict__ rlog, int* __restrict__ ridx,
                                float* __restrict__ masksOut)
{
    size_t m = (size_t)blockIdx.x * blockDim.x + threadIdx.x;
    if (m >= (size_t)M) return;
    const float* l = logits + m * 8;
    float mx = l[0];
    for (int j = 1; j < 8; ++j) mx = fmaxf(mx, l[j]);
    float e[8], s = 0.f;
    for (int j = 0; j < 8; ++j) { e[j] = expf(l[j] - mx); s += e[j]; }
    float best = -1.f; int bi = 0;
    for (int j = 0; j < 8; ++j) { float p = e[j] / s; if (p > best) { best = p; bi = j; } }
    rlog[m] = best; ridx[m] = bi;
    if (masksOut)
        for (int r = 0; r < 8; ++r) masksOut[(size_t)r * M + m] = (r == bi) ? 1.f : 0.f;
}

__global__ void region_sums_kernel(const float* __restrict__ rlog, const int* __restrict__ ridx,
                                   int N, float* __restrict__ sums, int* __restrict__ counts, int B)
{
    int t = threadIdx.x;
    if (t >= B * 8) return;
    int b = t / 8, r = t % 8;
    float s = 0.f; int cnt = 0;
    for (int n = 0; n < N; ++n) {
        size_t m = (size_t)b * N + n;
        if (ridx[m] == r) { s += rlog[m]; ++cnt; }
    }
    sums[t] = s; counts[t] = cnt;
}

__global__ void region_loss_kernel(const float* __restrict__ sums, int B, int N, float* lossOut)
{
    if (threadIdx.x || blockIdx.x) return;
    float tot = 0.f;
    for (int b = 0; b < B; ++b) {
        float sb = 0.f;
        for (int r = 0; r < 8; ++r) { float v = sums[b * 8 + r] / N; sb += v * v; }
        tot += sb;
    }
    lossOut[0] += 0.5f * (tot / B);
}

__global__ void h0_kernel(const float* __restrict__ b1, const float* __restrict__ W2,
                          const float* __restrict__ b2, float* __restrict__ h0, int C)
{
    int c = blockIdx.x * blockDim.x + threadIdx.x;
    if (c >= C) return;
    float s = b2[c];
    for (int j = 0; j < C; ++j) s += fmaxf(b1[j], 0.f) * W2[(size_t)j * C + c];
    h0[c] = fmaxf(s, 0.f);
}

__global__ void regvec_init_kernel(const int* __restrict__ counts, const float* __restrict__ h0,
                                   float* __restrict__ regvec, int B, int C, int N, size_t total)
{
    size_t t = (size_t)blockIdx.x * blockDim.x + threadIdx.x;
    if (t >= total) return;
    int c = (int)(t % C); size_t rb = t / C;
    int b = (int)(rb % B); int r = (int)(rb / B);
    regvec[t] = (counts[b * 8 + r] < N) ? h0[c] : 0.f;
}

__global__ void regmax_kernel(const float* __restrict__ H, const int* __restrict__ ridx,
                              float* __restrict__ regvec, int N, int C, int B, size_t total)
{
    size_t t = (size_t)blockIdx.x * blockDim.x + threadIdx.x;
    if (t >= total) return;
    int c = (int)(t % C); size_t m = t / C;
    int b = (int)(m / N);
    int r = ridx[m];
    // H >= 0 (outer relu) so fp32 bit pattern is order-preserving under uint max
    atomicMax((unsigned int*)&regvec[((size_t)r * B + b) * C + c], __float_as_uint(H[t]));
}

__global__ void region_concat_kernel(const float* __restrict__ x, const float* __restrict__ regvec,
                                     const int* __restrict__ ridx, const float* __restrict__ g,
                                     float* __restrict__ out, int N, int C, int B, size_t total)
{
    size_t t = (size_t)blockIdx.x * blockDim.x + threadIdx.x;
    if (t >= total) return;
    int C3 = 3 * C;
    int col = (int)(t % C3); size_t m = t / C3;
    int b = (int)(m / N);
    float v;
    if (col < C) v = x[m * C + col];
    else if (col < 2 * C) v = regvec[((size_t)ridx[m] * B + b) * C + (col - C)];
    else v = g[(size_t)b * C + (col - 2 * C)];
    out[t] = v;
}

// ===========================================================================
// Host side
// ===========================================================================
struct Lin { const float* b; const float* w; };
struct BNP { const float* b; const float* g; };
struct AttnP { Lin attn0, attn1; BNP attn_bn; Lin k, pos0, pos1; BNP pos_bn; Lin post_fc1, pre_fc1, pre_fc2, q, v; };

struct Arena {
    char* base; size_t off;
    float* alloc(size_t nfloats) {
        float* p = (float*)(base + off);
        off += ((nfloats * sizeof(float) + 255) & ~(size_t)255);
        return p;
    }
};

#define L1D(kern, total, ...) do { size_t _t = (total); \
    kern<<<dim3((unsigned)((_t + 255) / 256)), dim3(256), 0, st>>>(__VA_ARGS__); } while (0)

static void gemmN(hipStream_t st, const float* A, const Lin& W, int M, int N, int K,
                  float* out, bool relu, const float* res = nullptr)
{
    if (N <= 8) {   // pc heads (N=3), occ logits (N=8): wave-per-row GEMV
        gemm_smalln_kernel<<<dim3((M + 7) / 8), 256, 0, st>>>(A, W.w, W.b, res, out, M, N, K, relu ? 1 : 0);
        return;
    }
    dim3 g((N + 31) / 32, (M + 15) / 16);
    gemm_kernel<<<g, 32, 0, st>>>(A, W.w, W.b, res, out, M, N, K, relu ? 1 : 0);
}

static void gemm_attn0(hipStream_t st, const Lin& W, int M, int Nout, int K,
                       const float* kfeat, const float* q, const float* pos,
                       const int* idx, int Np, float* out)
{
    dim3 g(Nout / 16, M / 16);   // M = B*Np*16, Nout = 64: exact
    gemm_attn0_kernel<<<g, 32, 0, st>>>(W.w, W.b, out, Np, K, Nout, kfeat, q, pos, idx);
}

static void bn_relu(hipStream_t st, float* x, int M, int C, const BNP& p, float* stats)
{
    bn_stats_kernel<<<C, 256, 0, st>>>(x, M, C, stats);
    L1D(bn_apply_relu_kernel, (size_t)M * C, x, C, stats, p.g, p.b, (size_t)M * C);
}

static void run_grp(hipStream_t st, Arena& ar, const Lin& fc1, const Lin& fc2,
                    const float* x, const float* pc, int B, int N, int C, float* out)
{
    size_t mark = ar.off;
    int* idx = (int*)ar.alloc((size_t)B * N * 16);
    L1D(knn_kernel, (size_t)B * N, pc, pc, idx, B, N, N);
    int KA = C + 3, KApad = (KA + 3) & ~3, CH = C / 2;
    size_t sh = (size_t)(16 * KApad + 16 * CH) * sizeof(float);
    grp_kernel<<<B * N, 256, sh, st>>>(x, pc, idx, fc1.w, fc1.b, fc2.w, fc2.b, out, N, C);
    ar.off = mark;
}

static void run_attn(hipStream_t st, Arena& ar, const AttnP& P,
                     const float* decx, const float* decpc,
                     const float* encx, const float* encpc,
                     int B, int N, int Ne, int in1, int in2, float* out)
{
    const int D = 256;
    size_t mark = ar.off;
    int M = B * N, Me = B * Ne, Nc = N + Ne;
    size_t MK = (size_t)M * 16;
    float* dx    = ar.alloc((size_t)M * D);
    float* ex    = ar.alloc((size_t)Me * D);
    float* pcc   = ar.alloc((size_t)B * Nc * 3);
    float* xc    = ar.alloc((size_t)B * Nc * D);
    int*   fidx  = (int*)ar.alloc((size_t)B * N);
    float* fdist = ar.alloc((size_t)B * Nc);
    float* pcs   = ar.alloc((size_t)M * 3);
    float* xs    = ar.alloc((size_t)M * D);
    float* qb    = ar.alloc((size_t)M * D);
    float* kb    = ar.alloc((size_t)M * D);
    float* vb    = ar.alloc((size_t)M * D);
    int*   idx   = (int*)ar.alloc(MK);
    float* pdiff = ar.alloc(MK * 3);
    float* plin  = ar.alloc(MK * 64);
    float* stats = ar.alloc(128);
    float* pos   = ar.alloc(MK * D);
    float* av    = ar.alloc(MK * D);
    float* agg   = ar.alloc((size_t)M * D);
    float* alin  = plin;  // reuse: plin is dead after the pos1 GEMM

    gemmN(st, decx, P.pre_fc1, M, D, in1, dx, false);
    gemmN(st, encx, P.pre_fc2, Me, D, in2, ex, false);
    L1D(copy_rows_kernel, (size_t)B * N * 3,  decpc, pcc, N,  Nc, 0, 3, (size_t)B * N * 3);
    L1D(copy_rows_kernel, (size_t)B * Ne * 3, encpc, pcc, Ne, Nc, N, 3, (size_t)B * Ne * 3);
    L1D(copy_rows_kernel, (size_t)M * D,  dx, xc, N,  Nc, 0, D, (size_t)M * D);
    L1D(copy_rows_kernel, (size_t)Me * D, ex, xc, Ne, Nc, N, D, (size_t)Me * D);
    fps_kernel<<<B, 256, 0, st>>>(pcc, fidx, fdist, Nc, N);
    L1D(gather_rows_kernel, (size_t)M * 3, pcc, fidx, pcs, Nc, N, 3, (size_t)M * 3);
    L1D(gather_rows_kernel, (size_t)M * D, xc,  fidx, xs,  Nc, N, D, (size_t)M * D);
    gemmN(st, dx, P.q, M, D, D, qb, false);
    gemmN(st, xs, P.k, M, D, D, kb, false);
    gemmN(st, xs, P.v, M, D, D, vb, false);
    L1D(knn_kernel, (size_t)B * N, decpc, pcs, idx, B, N, N);
    L1D(pcdiff_kernel, MK * 3, decpc, pcs, idx, pdiff, N, N, MK * 3);
    gemmN(st, pdiff, P.pos0, (int)MK, 64, 3, plin, false);
    bn_relu(st, plin, (int)MK, 64, P.pos_bn, stats);
    gemmN(st, plin, P.pos1, (int)MK, D, 64, pos, false);
    gemm_attn0(st, P.attn0, (int)MK, 64, D, kb, qb, pos, idx, N, alin);
    bn_relu(st, alin, (int)MK, 64, P.attn_bn, stats);
    gemmN(st, alin, P.attn1, (int)MK, D, 64, av, false);
    L1D(softmax_k_kernel, (size_t)M * D, av, D, (size_t)M * D);
    L1D(agg_kernel, (size_t)M * D, vb, pos, av, idx, agg, N, D, (size_t)M * D);
    gemmN(st, agg, P.post_fc1, M, in1, D, out, false, decx);
    ar.off = mark;
}

static void run_region(hipStream_t st, Arena& ar, const Lin& occ, const Lin& fc1, const Lin& fc2,
                       const float* x, const float* g, int B, int N, int C,
                       float* outcat, float* lossOut, float* masksOut)
{
    size_t mark = ar.off;
    int M = B * N;
    float* logits = ar.alloc((size_t)M * 8);
    float* rlog   = ar.alloc((size_t)M);
    int*   ridx   = (int*)ar.alloc((size_t)M);
    float* sums   = ar.alloc(B * 8);
    int*   counts = (int*)ar.alloc(B * 8);
    float* Ht     = ar.alloc((size_t)M * C);
    float* H      = ar.alloc((size_t)M * C);
    float* h0     = ar.alloc(C);
    float* regvec = ar.alloc((size_t)8 * B * C);
    gemmN(st, x, occ, M, 8, C, logits, false);
    L1D(occ_top1_kernel, (size_t)M, logits, M, rlog, ridx, masksOut);
    region_sums_kernel<<<1, B * 8, 0, st>>>(rlog, ridx, N, sums, counts, B);
    region_loss_kernel<<<1, 1, 0, st>>>(sums, B, N, lossOut);
    gemmN(st, x, fc1, M, C, C, Ht, true);
    gemmN(st, Ht, fc2, M, C, C, H, true);
    h0_kernel<<<(C + 255) / 256, 256, 0, st>>>(fc1.b, fc2.w, fc2.b, h0, C);
    L1D(regvec_init_kernel, (size_t)8 * B * C, counts, h0, regvec, B, C, N, (size_t)8 * B * C);
    L1D(regmax_kernel, (size_t)M * C, H, ridx, regvec, N, C, B, (size_t)M * C);
    L1D(region_concat_kernel, (size_t)M * 3 * C, x, regvec, ridx, g, outcat, N, C, B, (size_t)M * 3 * C);
    ar.off = mark;
}

extern "C" void kernel_launch(void* const* d_in, const int* in_sizes, int n_in,
                              void* d_out, int out_size, void* d_ws, size_t ws_size,
                              hipStream_t stream)
{
    hipStream_t st = stream;
    const int B = in_sizes[0] / 512;                  // 8
    const int N3 = 128, N2 = 512, N1 = 2048;

    const float* gin     = (const float*)d_in[0];
    const float* enc_x1  = (const float*)d_in[1];
    const float* enc_pc1 = (const float*)d_in[2];
    const float* enc_x2  = (const float*)d_in[3];
    const float* enc_pc2 = (const float*)d_in[4];
    const float* enc_x3  = (const float*)d_in[5];
    const float* enc_pc3 = (const float*)d_in[6];

    // ---- params: jax sorted-dict flatten order; each Lin = (b, w), BN = (b, g)
    int pi = 7;
    auto L = [&](Lin& l) { l.b = (const float*)d_in[pi++]; l.w = (const float*)d_in[pi++]; };
    auto BNf = [&](BNP& b) { b.b = (const float*)d_in[pi++]; b.g = (const float*)d_in[pi++]; };
    auto AT = [&](AttnP& a) {
        L(a.attn0); L(a.attn1); BNf(a.attn_bn); L(a.k); L(a.pos0); L(a.pos1);
        BNf(a.pos_bn); L(a.post_fc1); L(a.pre_fc1); L(a.pre_fc2); L(a.q); L(a.v);
    };
    AttnP a1, a2, a3;  AT(a1); AT(a2); AT(a3);
    Lin g_fc1, g_fc2, g_fc3;                          L(g_fc1); L(g_fc2); L(g_fc3);
    Lin g_pc11a, g_pc11b, g_pc12a, g_pc12b, g_pc12c;  L(g_pc11a); L(g_pc11b); L(g_pc12a); L(g_pc12b); L(g_pc12c);
    Lin g_pc21, g_pc22a, g_pc22b, g_pc22c;            L(g_pc21); L(g_pc22a); L(g_pc22b); L(g_pc22c);
    Lin g_pc3;                                        L(g_pc3);
    Lin grp1_fc1, grp1_fc2;                           L(grp1_fc1); L(grp1_fc2);
    Lin grp2_fc1, grp2_fc2;                           L(grp2_fc1); L(grp2_fc2);
    Lin grp3_fc1, grp3_fc2;                           L(grp3_fc1); L(grp3_fc2);
    Lin post_fc1a, post_fc1b, post_fc1c;              L(post_fc1a); L(post_fc1b); L(post_fc1c);
    Lin post_fc2a, post_fc2b, post_fc2c;              L(post_fc2a); L(post_fc2b); L(post_fc2c);
    Lin post_fc3;                                     L(post_fc3);
    Lin pre_fc1a, pre_fc1b;                           L(pre_fc1a); L(pre_fc1b);
    Lin pre_fc2a, pre_fc2b;                           L(pre_fc2a); L(pre_fc2b);
    Lin pre_fc3a, pre_fc3b;                           L(pre_fc3a); L(pre_fc3b);
    Lin reg1_fc1, reg1_fc2, reg1_occ;                 L(reg1_fc1); L(reg1_fc2); L(reg1_occ);
    Lin reg2_fc1, reg2_fc2, reg2_occ;                 L(reg2_fc1); L(reg2_fc2); L(reg2_occ);

    // ---- outputs: pc3 | pc2 | pc1 | loss | masks(8,B,2048)
    float* out   = (float*)d_out;
    float* pc3o  = out;
    float* pc2o  = pc3o + (size_t)B * N3 * 3;
    float* pc1o  = pc2o + (size_t)B * N2 * 3;
    float* lossO = pc1o + (size_t)B * N1 * 3;
    float* masksO = lossO + 1;

    Arena ar{(char*)d_ws, 0};
    // persistent small buffers
    float* g1   = ar.alloc((size_t)B * 512);
    float* g2   = ar.alloc((size_t)B * 256);
    float* g3   = ar.alloc((size_t)B * 128);
    float* pc3a = ar.alloc((size_t)B * N3 * 3);
    float* g2b  = ar.alloc((size_t)B * 256);
    float* g1t  = ar.alloc((size_t)B * 512);
    float* g1b  = ar.alloc((size_t)B * 512);
    float* pc2a = ar.alloc((size_t)B * N2 * 3);
    float* pc1a = ar.alloc((size_t)B * N1 * 3);
    size_t base = ar.off;

    gemmN(st, gin, g_fc1, B, 512, 512, g1, true);
    gemmN(st, g1,  g_fc2, B, 256, 512, g2, true);
    gemmN(st, g2,  g_fc3, B, 128, 256, g3, true);
    gemmN(st, g3,  g_pc3, B, 384, 128, pc3a, false);

    hipMemsetAsync(lossO, 0, sizeof(float), st);

    // -------------------------- stage 3 (N=128, C=128) ----------------------
    {
        const int N = N3, C = 128, M = B * N;
        float* t    = ar.alloc((size_t)M * C);
        float* dec  = ar.alloc((size_t)M * C);
        float* decg = ar.alloc((size_t)M * C);
        float* deco = ar.alloc((size_t)M * C);
        gemmN(st, pc3a, pre_fc3a, M, C, 3, t, true);
        gemmN(st, t,    pre_fc3b, M, C, C, dec, false);
        run_grp(st, ar, grp3_fc1, grp3_fc2, dec, pc3a, B, N, C, decg);
        run_attn(st, ar, a3, decg, pc3a, enc_x3, enc_pc3, B, N, 128, C, 512, deco);
        gemmN(st, deco, post_fc3, M, 3, C, pc3o, false);
        ar.off = base;
    }

    // -------------------------- stage 2 (N=512, C=256) ----------------------
    {
        const int N = N2, C = 256, M = B * N;
        float* coarse = ar.alloc((size_t)M * 3);
        L1D(repeat3_kernel, (size_t)M * 3, pc3o, coarse, N3, 4, (size_t)M * 3);
        gemmN(st, g2, g_pc21, B, 256, 256, g2b, true);
        float* feat = ar.alloc((size_t)M * 261);
        L1D(build_feat_kernel, (size_t)M * 261, coarse, g2b, feat, N, 256, 0.2f, (size_t)M * 261);
        float* h1 = ar.alloc((size_t)M * C);
        float* h2 = ar.alloc((size_t)M * C);
        gemmN(st, feat, g_pc22a, M, C, 261, h1, true);
        gemmN(st, h1,   g_pc22b, M, C, C,   h2, true);
        gemmN(st, h2,   g_pc22c, M, 3, C,   pc2a, false, coarse);
        float* dec  = ar.alloc((size_t)M * C);
        gemmN(st, pc2a, pre_fc2a, M, C, 3, h1, true);   // h1 reused as temp
        gemmN(st, h1,   pre_fc2b, M, C, C, dec, false);
        float* decg = ar.alloc((size_t)M * C);
        run_grp(st, ar, grp2_fc1, grp2_fc2, dec, pc2a, B, N, C, decg);
        float* deco = ar.alloc((size_t)M * C);
        run_attn(st, ar, a2, decg, pc2a, enc_x2, enc_pc2, B, N, 512, 256, 256, deco);
        float* deccat = ar.alloc((size_t)M * 3 * C);
        run_region(st, ar, reg2_occ, reg2_fc1, reg2_fc2, deco, g2b, B, N, C, deccat, lossO, nullptr);
        float* p1 = ar.alloc((size_t)M * C);
        float* p2 = ar.alloc((size_t)M * C);
        gemmN(st, deccat, post_fc2a, M, C, 3 * C, p1, true);
        gemmN(st, p1,     post_fc2b, M, C, C,     p2, true);
        gemmN(st, p2,     post_fc2c, M, 3, C,     pc2o, false, pc2a);
        ar.off = base;
    }

    // -------------------------- stage 1 (N=2048, C=512) ---------------------
    {
        const int N = N1, C = 512, M = B * N;
        float* coarse = ar.alloc((size_t)M * 3);
        L1D(repeat3_kernel, (size_t)M * 3, pc2o, coarse, N2, 4, (size_t)M * 3);
        gemmN(st, g1,  g_pc11a, B, 512, 512, g1t, true);
        gemmN(st, g1t, g_pc11b, B, 512, 512, g1b, true);
        float* feat = ar.alloc((size_t)M * 517);
        L1D(build_feat_kernel, (size_t)M * 517, coarse, g1b, feat, N, 512, 0.05f, (size_t)M * 517);
        float* h1 = ar.alloc((size_t)M * C);
        float* h2 = ar.alloc((size_t)M * C);
        gemmN(st, feat, g_pc12a, M, C, 517, h1, true);
        gemmN(st, h1,   g_pc12b, M, C, C,   h2, true);
        gemmN(st, h2,   g_pc12c, M, 3, C,   pc1a, false, coarse);
        float* dec  = ar.alloc((size_t)M * C);
        gemmN(st, pc1a, pre_fc1a, M, C, 3, h1, true);
        gemmN(st, h1,   pre_fc1b, M, C, C, dec, false);
        float* decg = ar.alloc((size_t)M * C);
        run_grp(st, ar, grp1_fc1, grp1_fc2, dec, pc1a, B, N, C, decg);
        float* deco = ar.alloc((size_t)M * C);
        run_attn(st, ar, a1, decg, pc1a, enc_x1, enc_pc1, B, N, 2048, 512, 128, deco);
        float* deccat = ar.alloc((size_t)M * 3 * C);
        run_region(st, ar, reg1_occ, reg1_fc1, reg1_fc2, deco, g1b, B, N, C, deccat, lossO, masksO);
        float* p1 = ar.alloc((size_t)M * C);
        float* p2 = ar.alloc((size_t)M * C);
        gemmN(st, deccat, post_fc1a, M, C, 3 * C, p1, true);
        gemmN(st, p1,     post_fc1b, M, C, C,     p2, true);
        gemmN(st, p2,     post_fc1c, M, 3, C,     pc1o, false, pc1a);
        ar.off = base;
    }
    (void)n_in; (void)out_size; (void)ws_size;
}